// EdgeAttributePredictorConvNorm_36197984370744
// MI455X (gfx1250) — compile-verified
//
#include <hip/hip_runtime.h>
#include <hip/hip_bf16.h>

typedef __attribute__((ext_vector_type(2))) float v2f;
typedef __attribute__((ext_vector_type(8))) float v8f;
typedef __attribute__((ext_vector_type(4))) int   v4i;

#define N_NODES 50000
#define N_EDGES 400000
#define FDIM 128
#define NEG_SLOPE 0.1f
#define PN_EPS 1e-5f
#define M_TILES 3125            // N_NODES / 16

#if defined(__has_builtin)
#if __has_builtin(__builtin_amdgcn_global_load_async_to_lds_b128)
#define USE_ASYNC_LDS 1
#endif
#endif

typedef __attribute__((address_space(1))) v4i GV4;   // global int4*
typedef __attribute__((address_space(3))) v4i LV4;   // LDS int4*

__device__ __forceinline__ float leaky(float v) {
    return v >= 0.0f ? v : NEG_SLOPE * v;
}

__device__ __forceinline__ void atomic_add_f32(float* p, float v) {
    __hip_atomic_fetch_add(p, v, __ATOMIC_RELAXED, __HIP_MEMORY_SCOPE_AGENT);
}

__device__ __forceinline__ v8f wmma_f32(v2f a, v2f b, v8f c) {
    return __builtin_amdgcn_wmma_f32_16x16x4_f32(false, a, false, b,
                                                 (short)0, c, false, false);
}

// ---------------------------------------------------------------------------
// Utility: zero a float buffer
// ---------------------------------------------------------------------------
__global__ void zero_kernel(float* __restrict__ p, int n) {
    int i = blockIdx.x * blockDim.x + threadIdx.x;
    if (i < n) p[i] = 0.0f;
}

// ---------------------------------------------------------------------------
// Degree (message count per destination node)
// ---------------------------------------------------------------------------
__global__ void degree_kernel(const int* __restrict__ dst, float* __restrict__ cnt, int E) {
    int e = blockIdx.x * blockDim.x + threadIdx.x;
    if (e < E) atomic_add_f32(&cnt[dst[e]], 1.0f);
}

// ---------------------------------------------------------------------------
// Scatter-sum of messages h[src] into ssum[dst]; 1 thread = 4 features of 1 edge
// ---------------------------------------------------------------------------
__global__ void scatter_kernel(const float* __restrict__ h,
                               const int* __restrict__ src,
                               const int* __restrict__ dst,
                               float* __restrict__ ssum, int E) {
    int idx = blockIdx.x * blockDim.x + threadIdx.x;
    if (idx >= E * 32) return;
    int e  = idx >> 5;
    int f4 = (idx & 31) << 2;
    int s = src[e], d = dst[e];
    const float4 v = *(const float4*)(h + (size_t)s * FDIM + f4);
    float* base = ssum + (size_t)d * FDIM + f4;
    atomic_add_f32(base + 0, v.x);
    atomic_add_f32(base + 1, v.y);
    atomic_add_f32(base + 2, v.z);
    atomic_add_f32(base + 3, v.w);
}

// ---------------------------------------------------------------------------
// SAGE layer GEMM:  y = (ssum/max(cnt,1)) @ Wl + bl + h @ Wr
// One wave computes a full 16x128 output row-block: 8 accumulator tiles,
// A fragment loaded ONCE per K-step and reused across all 8 WMMAs.
// Two branch-free K loops (agg half, root half). PairNorm stats fused.
// ---------------------------------------------------------------------------
__global__ __launch_bounds__(256) void sage_gemm_kernel(
        const float* __restrict__ ssum, const float* __restrict__ cnt,
        const float* __restrict__ h,
        const float* __restrict__ Wl, const float* __restrict__ bl,
        const float* __restrict__ Wr,
        float* __restrict__ y, float* __restrict__ colsum, float* __restrict__ sumsq) {
    int wave = threadIdx.x >> 5;
    int lane = threadIdx.x & 31;
    int tileM = blockIdx.x * 8 + wave;      // wave-uniform
    if (tileM >= M_TILES) return;           // uniform exit: EXEC stays full
    int row = lane & 15;
    int hi  = lane >> 4;

    int m = tileM * 16 + row;               // A-matrix row for this lane
    float rc = 1.0f / fmaxf(cnt[m], 1.0f);
    const float* srow = ssum + (size_t)m * FDIM;
    const float* hrow = h    + (size_t)m * FDIM;

    v8f acc[8] = {};

    // ---- K half 1: mean-aggregated features through Wl ------------------
    for (int k = 0; k < 128; k += 4) {
        int k0 = k + 2 * hi;                // f32 A layout: hi half = K+2,K+3
        v2f a = *(const v2f*)(srow + k0);
        a.x *= rc; a.y *= rc;
        const float* w0 = Wl + (size_t)k0 * FDIM;
        __builtin_prefetch(w0 + 8 * FDIM, 0, 0);
#pragma unroll
        for (int nt = 0; nt < 8; ++nt) {
            int c = nt * 16 + row;
            v2f b; b.x = w0[c]; b.y = w0[FDIM + c];
            acc[nt] = wmma_f32(a, b, acc[nt]);
        }
    }
    // ---- K half 2: root features through Wr -----------------------------
    for (int k = 0; k < 128; k += 4) {
        int k0 = k + 2 * hi;
        v2f a = *(const v2f*)(hrow + k0);
        const float* w0 = Wr + (size_t)k0 * FDIM;
        __builtin_prefetch(w0 + 8 * FDIM, 0, 0);
#pragma unroll
        for (int nt = 0; nt < 8; ++nt) {
            int c = nt * 16 + row;
            v2f b; b.x = w0[c]; b.y = w0[FDIM + c];
            acc[nt] = wmma_f32(a, b, acc[nt]);
        }
    }

    // ---- epilogue: bias, store, PairNorm statistics ---------------------
    float lsq = 0.0f;
#pragma unroll
    for (int nt = 0; nt < 8; ++nt) {
        int c = nt * 16 + row;
        float bias = bl[c];
        float lsum = 0.0f;
#pragma unroll
        for (int i = 0; i < 8; ++i) {
            float v = acc[nt][i] + bias;    // D element (M = i + 8*hi, N = c)
            lsum += v;
            lsq  += v * v;
            int mr = tileM * 16 + i + 8 * hi;
            y[(size_t)mr * FDIM + c] = v;
        }
        atomic_add_f32(&colsum[c], lsum);
    }
#pragma unroll
    for (int off = 16; off > 0; off >>= 1) lsq += __shfl_xor(lsq, off, 32);
    if (lane == 0) atomic_add_f32(sumsq, lsq);
}

// ---------------------------------------------------------------------------
// PairNorm finalize: mu = colsum/N ; inv = rsqrt(eps + (sumsq - N*||mu||^2)/N)
// ---------------------------------------------------------------------------
__global__ void pairnorm_finalize_kernel(const float* __restrict__ colsum,
                                         const float* __restrict__ sumsq,
                                         float* __restrict__ mu,
                                         float* __restrict__ inv) {
    __shared__ float sh[128];
    int t = threadIdx.x;
    float m = colsum[t] / (float)N_NODES;
    mu[t] = m;
    sh[t] = m * m;
    __syncthreads();
    for (int s = 64; s > 0; s >>= 1) {
        if (t < s) sh[t] += sh[t + s];
        __syncthreads();
    }
    if (t == 0) {
        float centered = sumsq[0] - (float)N_NODES * sh[0];
        inv[0] = rsqrtf(PN_EPS + centered / (float)N_NODES);
    }
}

// ---------------------------------------------------------------------------
// PairNorm apply + LeakyReLU : h = leaky((y - mu[f]) * inv)
// ---------------------------------------------------------------------------
__global__ void pairnorm_apply_kernel(const float* __restrict__ y,
                                      const float* __restrict__ mu,
                                      const float* __restrict__ inv,
                                      float* __restrict__ hout, int total) {
    int i = blockIdx.x * blockDim.x + threadIdx.x;
    if (i >= total) return;
    int f = i & (FDIM - 1);
    hout[i] = leaky((y[i] - mu[f]) * inv[0]);
}

// ---------------------------------------------------------------------------
// Edge MLP: e = concat(h[src], h[dst]) (256) -> 128 -> 64 -> 8
// One wave per 16-edge tile; 4 waves / 128-thread block.
// Layer 1: 8 accumulators, A gathered once per K-step (src half + dst half).
// Intermediates staged in per-wave LDS (same-wave DS ops are in-order).
// W_out staged block-wide in LDS via async global->LDS when available.
// ---------------------------------------------------------------------------
__global__ __launch_bounds__(128) void edge_mlp_kernel(
        const float* __restrict__ h,
        const int* __restrict__ src, const int* __restrict__ dst,
        const float* __restrict__ Wf0, const float* __restrict__ bf0,
        const float* __restrict__ Wf1, const float* __restrict__ bf1,
        const float* __restrict__ Wo,  const float* __restrict__ bo,
        float* __restrict__ out) {
    __shared__ float e1[4][16 * 128];   // 32 KB (per-wave layer-1 activations)
    __shared__ float e2[4][16 * 64];    // 16 KB (per-wave layer-2 activations)
    __shared__ float wo_s[64 * 8];      //  2 KB (block-shared W_out)

    // ---- stage W_out into LDS (async DMA path when toolchain exposes it) --
#ifdef USE_ASYNC_LDS
    {
        float* g = (float*)Wo + threadIdx.x * 4;        // 128 thr * 4 = 512 f
        float* l = wo_s + threadIdx.x * 4;
        __builtin_amdgcn_global_load_async_to_lds_b128(
            (GV4*)(void*)g, (LV4*)(void*)l, 0, 0);
        asm volatile("s_wait_asynccnt 0" ::: "memory");
    }
#else
    *(float4*)(wo_s + threadIdx.x * 4) = *(const float4*)(Wo + threadIdx.x * 4);
#endif
    __syncthreads();

    int wave = threadIdx.x >> 5;
    int lane = threadIdx.x & 31;
    int tile = blockIdx.x * 4 + wave;   // 0..24999 (exact)
    int row = lane & 15;
    int hi  = lane >> 4;

    int e = tile * 16 + row;            // this lane's A-matrix edge row
    int s = src[e], d = dst[e];
    const float* sr = h + (size_t)s * FDIM;
    const float* dr = h + (size_t)d * FDIM;
    float* L1 = e1[wave];
    float* L2 = e2[wave];

    // ---- layer 1: [16x256] @ [256x128] -> leaky -> LDS ------------------
    {
        v8f acc[8] = {};
        for (int k = 0; k < 128; k += 4) {          // src half (K 0..127)
            int k0 = k + 2 * hi;
            v2f a = *(const v2f*)(sr + k0);
            const float* w0 = Wf0 + (size_t)k0 * 128;
            __builtin_prefetch(w0 + 8 * 128, 0, 0);
#pragma unroll
            for (int nt = 0; nt < 8; ++nt) {
                int c = nt * 16 + row;
                v2f b; b.x = w0[c]; b.y = w0[128 + c];
                acc[nt] = wmma_f32(a, b, acc[nt]);
            }
        }
        for (int k = 0; k < 128; k += 4) {          // dst half (K 128..255)
            int k0 = k + 2 * hi;
            v2f a = *(const v2f*)(dr + k0);
            const float* w0 = Wf0 + (size_t)(128 + k0) * 128;
            __builtin_prefetch(w0 + 8 * 128, 0, 0);
#pragma unroll
            for (int nt = 0; nt < 8; ++nt) {
                int c = nt * 16 + row;
                v2f b; b.x = w0[c]; b.y = w0[128 + c];
                acc[nt] = wmma_f32(a, b, acc[nt]);
            }
        }
#pragma unroll
        for (int nt = 0; nt < 8; ++nt) {
            int c = nt * 16 + row;
            float bias = bf0[c];
#pragma unroll
            for (int i = 0; i < 8; ++i)
                L1[(i + 8 * hi) * 128 + c] = leaky(acc[nt][i] + bias);
        }
    }

    // ---- layer 2: [16x128] @ [128x64] -> leaky -> LDS -------------------
    {
        v8f acc[4] = {};
        for (int k = 0; k < 128; k += 4) {
            int k0 = k + 2 * hi;
            v2f a = *(const v2f*)(L1 + row * 128 + k0);
            const float* w0 = Wf1 + (size_t)k0 * 64;
#pragma unroll
            for (int nt = 0; nt < 4; ++nt) {
                int c = nt * 16 + row;
                v2f b; b.x = w0[c]; b.y = w0[64 + c];
                acc[nt] = wmma_f32(a, b, acc[nt]);
            }
        }
#pragma unroll
        for (int nt = 0; nt < 4; ++nt) {
            int c = nt * 16 + row;
            float bias = bf1[c];
#pragma unroll
            for (int i = 0; i < 8; ++i)
                L2[(i + 8 * hi) * 64 + c] = leaky(acc[nt][i] + bias);
        }
    }

    // ---- layer 3: [16x64] @ [64x8] (B zero-padded to 16 cols, from LDS) --
    {
        int ncol = row;                 // 0..15; only 0..7 are real outputs
        v8f acc = {};
        for (int k = 0; k < 64; k += 4) {
            int k0 = k + 2 * hi;
            v2f a = *(const v2f*)(L2 + row * 64 + k0);
            v2f b;
            b.x = (ncol < 8) ? wo_s[k0 * 8 + ncol] : 0.0f;
            b.y = (ncol < 8) ? wo_s[(k0 + 1) * 8 + ncol] : 0.0f;
            acc = wmma_f32(a, b, acc);
        }
        if (ncol < 8) {
            float bias = bo[ncol];
#pragma unroll
            for (int i = 0; i < 8; ++i) {
                int m = tile * 16 + i + 8 * hi;
                out[(size_t)m * 8 + ncol] = acc[i] + bias;
            }
        }
    }
}

// ---------------------------------------------------------------------------
// Host launcher
// ---------------------------------------------------------------------------
extern "C" void kernel_launch(void* const* d_in, const int* in_sizes, int n_in,
                              void* d_out, int out_size, void* d_ws, size_t ws_size,
                              hipStream_t stream) {
    const float* x   = (const float*)d_in[0];
    const int* eidx  = (const int*)d_in[1];
    const int* src   = eidx;
    const int* dst   = eidx + N_EDGES;
    const float* Wl0 = (const float*)d_in[2];
    const float* bl0 = (const float*)d_in[3];
    const float* Wr0 = (const float*)d_in[4];
    const float* Wl1 = (const float*)d_in[5];
    const float* bl1 = (const float*)d_in[6];
    const float* Wr1 = (const float*)d_in[7];
    const float* Wf0 = (const float*)d_in[8];
    const float* bf0 = (const float*)d_in[9];
    const float* Wf1 = (const float*)d_in[10];
    const float* bf1 = (const float*)d_in[11];
    const float* Wo  = (const float*)d_in[12];
    const float* bo  = (const float*)d_in[13];
    float* out = (float*)d_out;

    float* ws = (float*)d_ws;
    size_t off = 0;
    float* cnt    = ws + off; off += N_NODES;
    float* ssum   = ws + off; off += (size_t)N_NODES * FDIM;
    float* ybuf   = ws + off; off += (size_t)N_NODES * FDIM;
    float* hbuf   = ws + off; off += (size_t)N_NODES * FDIM;
    float* colsum = ws + off; off += 128;
    float* sumsq  = ws + off; off += 1;
    float* mu     = ws + off; off += 128;
    float* inv    = ws + off; off += 1;

    const int ZT = 256;
    int nzero0 = N_NODES + N_NODES * FDIM;          // cnt + ssum (contiguous)
    int nstats = 130;                               // colsum + sumsq (+slack)
    int ntot   = N_NODES * FDIM;
    int sage_blocks = (M_TILES + 7) / 8;            // 8 waves / block

    // ---- layer 0 ----
    zero_kernel<<<(nzero0 + ZT - 1) / ZT, ZT, 0, stream>>>(cnt, nzero0);
    zero_kernel<<<1, ZT, 0, stream>>>(colsum, nstats);
    degree_kernel<<<(N_EDGES + ZT - 1) / ZT, ZT, 0, stream>>>(dst, cnt, N_EDGES);
    scatter_kernel<<<(N_EDGES * 32 + ZT - 1) / ZT, ZT, 0, stream>>>(x, src, dst, ssum, N_EDGES);
    sage_gemm_kernel<<<sage_blocks, 256, 0, stream>>>(ssum, cnt, x, Wl0, bl0, Wr0,
                                                      ybuf, colsum, sumsq);
    pairnorm_finalize_kernel<<<1, 128, 0, stream>>>(colsum, sumsq, mu, inv);
    pairnorm_apply_kernel<<<(ntot + ZT - 1) / ZT, ZT, 0, stream>>>(ybuf, mu, inv, hbuf, ntot);

    // ---- layer 1 ----
    zero_kernel<<<(ntot + ZT - 1) / ZT, ZT, 0, stream>>>(ssum, ntot);
    zero_kernel<<<1, ZT, 0, stream>>>(colsum, nstats);
    scatter_kernel<<<(N_EDGES * 32 + ZT - 1) / ZT, ZT, 0, stream>>>(hbuf, src, dst, ssum, N_EDGES);
    sage_gemm_kernel<<<sage_blocks, 256, 0, stream>>>(ssum, cnt, hbuf, Wl1, bl1, Wr1,
                                                      ybuf, colsum, sumsq);
    pairnorm_finalize_kernel<<<1, 128, 0, stream>>>(colsum, sumsq, mu, inv);
    pairnorm_apply_kernel<<<(ntot + ZT - 1) / ZT, ZT, 0, stream>>>(ybuf, mu, inv, hbuf, ntot);

    // ---- edge MLP ----
    edge_mlp_kernel<<<N_EDGES / 16 / 4, 128, 0, stream>>>(hbuf, src, dst,
                                                          Wf0, bf0, Wf1, bf1, Wo, bo, out);
}